// VGAE_83021717831897
// MI455X (gfx1250) — compile-verified
//
#include <hip/hip_runtime.h>
#include <hip/hip_bf16.h>
#include <math.h>

// ---------------- problem constants ----------------
#define NTOT   1025
#define NN     1024
#define INCH   256
#define HID    128
#define HEADS  4
#define H1C    (HEADS*HID)   // 512
#define LAT    64
#define ECNT   16384
#define ETOT   (ECNT + NTOT) // 17409 (self loops appended)
#define NEG_SLOPE 0.2f

typedef __attribute__((ext_vector_type(2))) float v2f;
typedef __attribute__((ext_vector_type(8))) float v8f;

// ---------------- f32 WMMA GEMM: C[M,N] = A[M,K] @ B[K,N] (row major) ------
// One wave32 per 16x16 C tile, V_WMMA_F32_16X16X4_F32, K stepped by 4.
// A frag layout (ISA 7.12.2, 32-bit A 16x4): lane l<16 : M=l, VGPR0=K0,VGPR1=K1
//                                            lane l>=16: M=l-16, VGPR0=K2,VGPR1=K3
// B frag (row striped across lanes): VGPR0 = B[K0][n0+l] (lanes0-15) / B[K2] (16-31)
// C/D: VGPR r -> row m0 + r + 8*(lane/16), col n0 + lane%16.
__global__ __launch_bounds__(32)
void gemm_wmma_f32(const float* __restrict__ A, const float* __restrict__ B,
                   float* __restrict__ C, int M, int N, int K) {
    const int lane = threadIdx.x & 31;
    const int half = lane >> 4;        // 0 or 1
    const int l    = lane & 15;
    const int m0   = blockIdx.y * 16;
    const int n0   = blockIdx.x * 16;

    int rowA = m0 + l;                 // clamp ragged M for loads (stores guarded)
    if (rowA >= M) rowA = M - 1;
    const int col = n0 + l;            // N is always a multiple of 16 here

    const float* __restrict__ Ap = A + (size_t)rowA * K + 2 * half;
    const float* __restrict__ Bp = B + (size_t)(2 * half) * N + col;

    v8f acc = {};
    for (int k = 0; k < K; k += 4) {
        __builtin_prefetch(Ap + k + 64, 0, 0);   // global_prefetch_b8
        v2f a = { Ap[k], Ap[k + 1] };
        v2f b = { Bp[(size_t)k * N], Bp[(size_t)(k + 1) * N] };
        acc = __builtin_amdgcn_wmma_f32_16x16x4_f32(
            /*neg_a=*/false, a, /*neg_b=*/false, b,
            /*c_mod=*/(short)0, acc, /*reuse_a=*/false, /*reuse_b=*/false);
    }
#pragma unroll
    for (int r = 0; r < 8; ++r) {
        int row = m0 + r + 8 * half;
        if (row < M) C[(size_t)row * N + col] = acc[r];
    }
}

// ---------------- attention score dots: al = sum_c h[n,hd,c]*a[hd,c] -------
__global__ void att_dots(const float* __restrict__ h, const float* __restrict__ asrc,
                         const float* __restrict__ adst, float* __restrict__ als,
                         float* __restrict__ ald, int n, int heads, int ch) {
    int i = blockIdx.x * blockDim.x + threadIdx.x;
    if (i >= n * heads) return;
    int node = i / heads, hd = i % heads;
    const float* hp = h + (size_t)node * heads * ch + (size_t)hd * ch;
    const float* as = asrc + (size_t)hd * ch;
    const float* ad = adst + (size_t)hd * ch;
    float s0 = 0.f, s1 = 0.f;
    for (int c = 0; c < ch; ++c) { float v = hp[c]; s0 += v * as[c]; s1 += v * ad[c]; }
    als[i] = s0; ald[i] = s1;
}

// ---------------- fill ----------------
__global__ void fill_f32(float* __restrict__ p, float v, int n) {
    int i = blockIdx.x * blockDim.x + threadIdx.x;
    if (i < n) p[i] = v;
}

__device__ __forceinline__ void atomicMaxFloat(float* addr, float v) {
    if (v >= 0.f) atomicMax((int*)addr, __float_as_int(v));
    else          atomicMin((unsigned int*)addr, __float_as_uint(v));
}

__device__ __forceinline__ void edge_ids(int edge, const int* __restrict__ ei,
                                         int& src, int& dst) {
    if (edge < ECNT) { src = ei[edge]; dst = ei[ECNT + edge]; }
    else             { src = edge - ECNT; dst = edge - ECNT; }  // self loops
}

// ---------------- pass 1: leaky-relu score + segment max -------------------
__global__ void edge_max(const float* __restrict__ als, const float* __restrict__ ald,
                         const int* __restrict__ ei, float* __restrict__ eraw,
                         float* __restrict__ m, int heads) {
    int i = blockIdx.x * blockDim.x + threadIdx.x;
    if (i >= ETOT * heads) return;
    int edge = i / heads, hd = i % heads;
    int src, dst; edge_ids(edge, ei, src, dst);
    float t = als[src * heads + hd] + ald[dst * heads + hd];
    t = (t > 0.f) ? t : NEG_SLOPE * t;
    eraw[i] = t;
    atomicMaxFloat(&m[dst * heads + hd], t);
}

// ---------------- pass 2: exp(e - m[dst]) + segment sum --------------------
__global__ void edge_exp(float* __restrict__ e, const float* __restrict__ m,
                         const int* __restrict__ ei, float* __restrict__ s, int heads) {
    int i = blockIdx.x * blockDim.x + threadIdx.x;
    if (i >= ETOT * heads) return;
    int edge = i / heads, hd = i % heads;
    int src, dst; edge_ids(edge, ei, src, dst);
    float v = expf(e[i] - m[dst * heads + hd]);
    e[i] = v;
    atomicAdd(&s[dst * heads + hd], v);
}

// ---------------- pass 3: agg[dst] += (e/s[dst]) * h[src] ------------------
__global__ void edge_agg(const float* __restrict__ e, const float* __restrict__ s,
                         const int* __restrict__ ei, const float* __restrict__ h,
                         float* __restrict__ agg, int heads, int ch) {
    long i = (long)blockIdx.x * blockDim.x + threadIdx.x;
    long total = (long)ETOT * heads * ch;
    if (i >= total) return;
    int c = (int)(i % (heads * ch));
    int edge = (int)(i / (heads * ch));
    int hd = c / ch;
    int src, dst; edge_ids(edge, ei, src, dst);
    float alpha = e[edge * heads + hd] / s[dst * heads + hd];
    atomicAdd(&agg[(size_t)dst * heads * ch + c],
              alpha * h[(size_t)src * heads * ch + c]);
}

// ---------------- bias (+ optional ELU), in place --------------------------
__global__ void bias_act(float* __restrict__ a, const float* __restrict__ b,
                         int n, int c, int elu) {
    int i = blockIdx.x * blockDim.x + threadIdx.x;
    if (i >= n * c) return;
    float v = a[i] + b[i % c];
    if (elu) v = (v > 0.f) ? v : (expf(v) - 1.f);
    a[i] = v;
}

// ---------------- ht = z[task] @ Wt + b1  (128 outs, K=64) -----------------
__global__ __launch_bounds__(128)
void ht_kernel(const float* __restrict__ z, const float* __restrict__ Wt,
               const float* __restrict__ b1, const int* __restrict__ task,
               float* __restrict__ ht) {
    int t = threadIdx.x;
    const float* zp = z + (size_t)(*task) * LAT;
    float s = b1[t];
#pragma unroll
    for (int k = 0; k < LAT; ++k) s += zp[k] * Wt[k * HID + t];
    ht[t] = s;
}

// ---------------- pairwise decoder: 1024x1024 sigmoid ----------------------
__global__ __launch_bounds__(256)
void pairwise(const float* __restrict__ hi, const float* __restrict__ hj,
              const float* __restrict__ ht, const float* __restrict__ w2,
              const float* __restrict__ b2, float* __restrict__ out) {
    __shared__ float shi[16 * HID];
    __shared__ float shj[16 * HID];
    __shared__ float sht[HID];
    __shared__ float sw2[HID];
    const int bi = blockIdx.y, bj = blockIdx.x, t = threadIdx.x;
    for (int k = t; k < 16 * HID; k += 256) {
        shi[k] = hi[(size_t)bi * 16 * HID + k];
        shj[k] = hj[(size_t)bj * 16 * HID + k];
    }
    if (t < HID) { sht[t] = ht[t]; sw2[t] = w2[t]; }
    __syncthreads();
    const int ti = t >> 4, tj = t & 15;
    const float* pi = &shi[ti * HID];
    const float* pj = &shj[tj * HID];
    float acc = b2[0];
#pragma unroll 8
    for (int h = 0; h < HID; ++h)
        acc += fmaxf(pi[h] + pj[h] + sht[h], 0.f) * sw2[h];
    out[(size_t)(bi * 16 + ti) * NN + (bj * 16 + tj)] = 1.f / (1.f + expf(-acc));
}

// ---------------- host side ----------------
static inline int cdiv(long a, long b) { return (int)((a + b - 1) / b); }

extern "C" void kernel_launch(void* const* d_in, const int* in_sizes, int n_in,
                              void* d_out, int out_size, void* d_ws, size_t ws_size,
                              hipStream_t stream) {
    const float* x     = (const float*)d_in[0];
    const float* W1    = (const float*)d_in[1];
    const float* asrc1 = (const float*)d_in[2];
    const float* adst1 = (const float*)d_in[3];
    const float* b1g   = (const float*)d_in[4];
    const float* muW   = (const float*)d_in[5];
    const float* muas  = (const float*)d_in[6];
    const float* muad  = (const float*)d_in[7];
    const float* mub   = (const float*)d_in[8];
    const float* decW1 = (const float*)d_in[13];
    const float* decb1 = (const float*)d_in[14];
    const float* decW2 = (const float*)d_in[15];
    const float* decb2 = (const float*)d_in[16];
    const int*   ei    = (const int*)d_in[17];
    const int*   task  = (const int*)d_in[18];
    float* out = (float*)d_out;

    // workspace carve-up (floats)
    float* w = (float*)d_ws;
    size_t off = 0;
    float* h1   = w + off; off += (size_t)NTOT * H1C;   // x@W1
    float* agg1 = w + off; off += (size_t)NTOT * H1C;   // layer1 agg -> h (ELU in place)
    float* al1s = w + off; off += (size_t)NTOT * HEADS;
    float* al1d = w + off; off += (size_t)NTOT * HEADS;
    float* m1   = w + off; off += (size_t)NTOT * HEADS;
    float* s1   = w + off; off += (size_t)NTOT * HEADS;
    float* e1   = w + off; off += (size_t)ETOT * HEADS;
    float* muh  = w + off; off += (size_t)NTOT * LAT;
    float* al2s = w + off; off += NTOT;
    float* al2d = w + off; off += NTOT;
    float* m2   = w + off; off += NTOT;
    float* s2   = w + off; off += NTOT;
    float* e2   = w + off; off += (size_t)ETOT;
    float* z    = w + off; off += (size_t)NTOT * LAT;   // layer2 agg -> z
    float* hi   = w + off; off += (size_t)NN * HID;
    float* hj   = w + off; off += (size_t)NN * HID;
    float* htb  = w + off; off += HID;

    const int TB = 256;

    // ---- GAT layer 1 ----
    gemm_wmma_f32<<<dim3(H1C / 16, cdiv(NTOT, 16)), 32, 0, stream>>>(
        x, W1, h1, NTOT, H1C, INCH);
    att_dots<<<cdiv(NTOT * HEADS, TB), TB, 0, stream>>>(
        h1, asrc1, adst1, al1s, al1d, NTOT, HEADS, HID);
    fill_f32<<<cdiv(NTOT * HEADS, TB), TB, 0, stream>>>(m1, -INFINITY, NTOT * HEADS);
    fill_f32<<<cdiv(NTOT * HEADS, TB), TB, 0, stream>>>(s1, 0.f, NTOT * HEADS);
    fill_f32<<<cdiv((long)NTOT * H1C, TB), TB, 0, stream>>>(agg1, 0.f, NTOT * H1C);
    edge_max<<<cdiv((long)ETOT * HEADS, TB), TB, 0, stream>>>(al1s, al1d, ei, e1, m1, HEADS);
    edge_exp<<<cdiv((long)ETOT * HEADS, TB), TB, 0, stream>>>(e1, m1, ei, s1, HEADS);
    edge_agg<<<cdiv((long)ETOT * HEADS * HID, TB), TB, 0, stream>>>(
        e1, s1, ei, h1, agg1, HEADS, HID);
    bias_act<<<cdiv((long)NTOT * H1C, TB), TB, 0, stream>>>(agg1, b1g, NTOT, H1C, 1);

    // ---- GAT layer 2 (mu head, heads=1, ch=LAT) ----
    gemm_wmma_f32<<<dim3(LAT / 16, cdiv(NTOT, 16)), 32, 0, stream>>>(
        agg1, muW, muh, NTOT, LAT, H1C);
    att_dots<<<cdiv(NTOT, TB), TB, 0, stream>>>(muh, muas, muad, al2s, al2d, NTOT, 1, LAT);
    fill_f32<<<cdiv(NTOT, TB), TB, 0, stream>>>(m2, -INFINITY, NTOT);
    fill_f32<<<cdiv(NTOT, TB), TB, 0, stream>>>(s2, 0.f, NTOT);
    fill_f32<<<cdiv((long)NTOT * LAT, TB), TB, 0, stream>>>(z, 0.f, NTOT * LAT);
    edge_max<<<cdiv((long)ETOT, TB), TB, 0, stream>>>(al2s, al2d, ei, e2, m2, 1);
    edge_exp<<<cdiv((long)ETOT, TB), TB, 0, stream>>>(e2, m2, ei, s2, 1);
    edge_agg<<<cdiv((long)ETOT * LAT, TB), TB, 0, stream>>>(e2, s2, ei, muh, z, 1, LAT);
    bias_act<<<cdiv((long)NTOT * LAT, TB), TB, 0, stream>>>(z, mub, NTOT, LAT, 0);

    // ---- decoder ----
    gemm_wmma_f32<<<dim3(HID / 16, NN / 16), 32, 0, stream>>>(
        z, decW1, hi, NN, HID, LAT);                       // Wi = dec_W1[0:64]
    gemm_wmma_f32<<<dim3(HID / 16, NN / 16), 32, 0, stream>>>(
        z, decW1 + (size_t)LAT * HID, hj, NN, HID, LAT);   // Wj = dec_W1[64:128]
    ht_kernel<<<1, 128, 0, stream>>>(z, decW1 + (size_t)2 * LAT * HID, decb1, task, htb);
    pairwise<<<dim3(NN / 16, NN / 16), 256, 0, stream>>>(hi, hj, htb, decW2, decb2, out);
    (void)in_sizes; (void)n_in; (void)out_size; (void)ws_size;
}